// CartesianMACE_52510270161489
// MI455X (gfx1250) — compile-verified
//
#include <hip/hip_runtime.h>

// CDNA5 (gfx1250, wave32) kernel for the CartesianMACE reference.
//
// Only the rank-0 (tail=1) channel feeds the final (1,2) output:
//   t = cw0[0][n] @ h0[n] + mw0[0][n] @ msg0_r0[n]        (16x16 matvecs)
//   y = cw1[0][n] @ t     + mw1[0][n] @ msg1_r0[n]
//   s[n] = sum_j y[j];  out[k] = sum_n s[n]*w_pred[k][n] + b_pred[k]
//
// ~215 MB streamed once -> HBM-bound (~9 us at 23.3 TB/s). One wave per node;
// matvecs via chained V_WMMA_F32_16X16X4_F32 (B = vector replicated across all
// 16 columns); weight loads are coalesced b64 matching the WMMA A-layout with
// TH_LOAD_NT (read-once, don't pollute the 192MB L2). Per-node vectors and
// w_pred are wave-uniform -> readfirstlane forces them onto the SMEM path.
// Cross-lane broadcasts use v_readlane (no LDS traffic) instead of bpermute.

typedef float v2f __attribute__((ext_vector_type(2)));
typedef float v8f __attribute__((ext_vector_type(8)));

static __device__ __forceinline__ v8f wmma4(v2f a, v2f b, v8f c) {
  // 8 args: (neg_a, A, neg_b, B, c_mod, C, reuse_a, reuse_b)
  return __builtin_amdgcn_wmma_f32_16x16x4_f32(false, a, false, b, (short)0, c,
                                               false, false);
}

static __device__ __forceinline__ float bcast_lane(float v, int srcLane) {
  return __int_as_float(__builtin_amdgcn_readlane(__float_as_int(v), srcLane));
}

// Uniform 16-float vector -> scalar registers (address is wave-uniform).
static __device__ __forceinline__ void load_uniform16(
    const float* __restrict__ x, float xv[16]) {
#pragma unroll
  for (int i = 0; i < 16; ++i) xv[i] = x[i];
}

// Build the per-lane B operand from a uniform 16-float vector, replicated
// across all 16 columns. Chunk kk covers K = 4kk..4kk+3; lanes 0-15 need
// {x[4kk], x[4kk+1]}, lanes 16-31 need {x[4kk+2], x[4kk+3]}.
static __device__ __forceinline__ void make_bvec(const float xv[16], int hi,
                                                 v2f bx[4]) {
#pragma unroll
  for (int kk = 0; kk < 4; ++kk) {
    bx[kk].x = hi ? xv[4 * kk + 2] : xv[4 * kk + 0];
    bx[kk].y = hi ? xv[4 * kk + 3] : xv[4 * kk + 1];
  }
}

// c += W @ x via 4 chained 16x16x4 WMMAs. W is one node's row-major 16x16.
// A-layout: lane (r,hi) loads float2 at element offset r*16 + 4*kk + 2*hi.
static __device__ __forceinline__ v8f matvec_acc(const float* __restrict__ W,
                                                 const v2f bx[4], int r, int hi,
                                                 v8f c) {
  const v2f* Wp = (const v2f*)W;
  const int base = r * 8 + hi;
  v2f a[4];
#pragma unroll
  for (int kk = 0; kk < 4; ++kk)
    a[kk] = __builtin_nontemporal_load(Wp + base + 2 * kk);
#pragma unroll
  for (int kk = 0; kk < 4; ++kk) c = wmma4(a[kk], bx[kk], c);
  return c;
}

__global__ __launch_bounds__(256) void mace_rank0_kernel(
    const float* __restrict__ h0, const float* __restrict__ cw0,
    const float* __restrict__ mw0, const float* __restrict__ cw1,
    const float* __restrict__ mw1, const float* __restrict__ msg0,
    const float* __restrict__ msg1, const float* __restrict__ w_pred,
    float* __restrict__ partials, int N) {
  const int lane = threadIdx.x & 31;
  const int wave = threadIdx.x >> 5;
  const int r = lane & 15;   // matrix row handled by this lane
  const int hi = lane >> 4;  // K-pair selector (0: k={0,1}, 1: k={2,3})

  const int wavesPerBlock = blockDim.x >> 5;
  const int gwave = blockIdx.x * wavesPerBlock + wave;
  const int totalWaves = gridDim.x * wavesPerBlock;

  float acc0 = 0.f, acc1 = 0.f;

  for (int n0 = gwave; n0 < N; n0 += totalWaves) {  // wave-uniform loop
    // n0 is wave-uniform in fact; make it provably uniform for the compiler
    // so vector/w_pred reads become scalar (SMEM) loads.
    const int n = __builtin_amdgcn_readfirstlane(n0);
    const size_t mOff = (size_t)n * 256;
    const size_t vOff = (size_t)n * 16;

    float xh[16], xm0[16], xm1[16];
    load_uniform16(h0 + vOff, xh);
    load_uniform16(msg0 + vOff, xm0);
    load_uniform16(msg1 + vOff, xm1);

    v2f bh[4], bm0[4], bm1[4];
    make_bvec(xh, hi, bh);
    make_bvec(xm0, hi, bm0);
    make_bvec(xm1, hi, bm1);

    // ---- layer 0: t = cw0[n] @ h0[n] + mw0[n] @ msg0[n] ----
    v8f c0 = {};
    c0 = matvec_acc(cw0 + mOff, bh, r, hi, c0);
    c0 = matvec_acc(mw0 + mOff, bm0, r, hi, c0);

    // Rebuild B operand for layer 1 from D (all 16 columns identical):
    // t[m] (m=0..7) lives in c0[m] on lane 0; t[8+m] in c0[m] on lane 16.
    float t16[16];
#pragma unroll
    for (int m = 0; m < 8; ++m) {
      t16[m] = bcast_lane(c0[m], 0);
      t16[8 + m] = bcast_lane(c0[m], 16);
    }
    v2f bt[4];
    make_bvec(t16, hi, bt);

    // ---- layer 1: y = cw1[n] @ t + mw1[n] @ msg1[n] ----
    v8f c1 = {};
    c1 = matvec_acc(cw1 + mOff, bt, r, hi, c1);
    c1 = matvec_acc(mw1 + mOff, bm1, r, hi, c1);

    // s = sum_j y[j]; lanes<16 hold rows 0-7 of a column, lanes>=16 rows 8-15.
    float p = c1[0] + c1[1] + c1[2] + c1[3] + c1[4] + c1[5] + c1[6] + c1[7];
    float s = bcast_lane(p, 0) + bcast_lane(p, 16);  // wave-uniform

    acc0 += s * w_pred[n];      // uniform scalar math
    acc1 += s * w_pred[N + n];
  }

  // block reduction over waves (deterministic, no float atomics)
  __shared__ float wacc[16][2];
  if (lane == 0) {
    wacc[wave][0] = acc0;
    wacc[wave][1] = acc1;
  }
  __syncthreads();
  if (threadIdx.x == 0) {
    float b0 = 0.f, b1 = 0.f;
    for (int w = 0; w < wavesPerBlock; ++w) {
      b0 += wacc[w][0];
      b1 += wacc[w][1];
    }
    partials[2 * (size_t)blockIdx.x + 0] = b0;
    partials[2 * (size_t)blockIdx.x + 1] = b1;
  }
}

__global__ __launch_bounds__(256) void mace_reduce_kernel(
    const float* __restrict__ partials, int numPartials,
    const float* __restrict__ b_pred, float* __restrict__ out) {
  __shared__ float s0[256], s1[256];
  float a0 = 0.f, a1 = 0.f;
  for (int i = threadIdx.x; i < numPartials; i += blockDim.x) {
    a0 += partials[2 * (size_t)i + 0];
    a1 += partials[2 * (size_t)i + 1];
  }
  s0[threadIdx.x] = a0;
  s1[threadIdx.x] = a1;
  __syncthreads();
  for (int off = 128; off > 0; off >>= 1) {
    if ((int)threadIdx.x < off) {
      s0[threadIdx.x] += s0[threadIdx.x + off];
      s1[threadIdx.x] += s1[threadIdx.x + off];
    }
    __syncthreads();
  }
  if (threadIdx.x == 0) {
    out[0] = s0[0] + b_pred[0];
    out[1] = s1[0] + b_pred[1];
  }
}

extern "C" void kernel_launch(void* const* d_in, const int* in_sizes, int n_in,
                              void* d_out, int out_size, void* d_ws,
                              size_t ws_size, hipStream_t stream) {
  (void)n_in;
  (void)out_size;
  const int N = in_sizes[0] / 16;  // h0 is (N,16,1)

  const float* h0 = (const float*)d_in[0];
  const float* cw0 = (const float*)d_in[1];  // (3,N,16,16): [0] = first N*256
  const float* mw0 = (const float*)d_in[2];
  const float* cw1 = (const float*)d_in[3];
  const float* mw1 = (const float*)d_in[4];
  const float* msg0_r0 = (const float*)d_in[5];
  const float* msg1_r0 = (const float*)d_in[8];
  const float* w_pred = (const float*)d_in[11];  // (2,N)
  const float* b_pred = (const float*)d_in[12];  // (2,)
  float* partials = (float*)d_ws;
  float* out = (float*)d_out;

  const int threads = 256;
  const int wavesPerBlock = threads / 32;
  int blocks = (N + wavesPerBlock - 1) / wavesPerBlock;  // 1 node per wave
  // keep per-block partials inside the workspace
  const size_t maxBlocks = ws_size / (2 * sizeof(float));
  if ((size_t)blocks > maxBlocks) blocks = (int)maxBlocks;
  if (blocks < 1) blocks = 1;

  mace_rank0_kernel<<<blocks, threads, 0, stream>>>(
      h0, cw0, mw0, cw1, mw1, msg0_r0, msg1_r0, w_pred, partials, N);
  mace_reduce_kernel<<<1, threads, 0, stream>>>(partials, blocks, b_pred, out);
}